// PoseCNN_75462575390933
// MI455X (gfx1250) — compile-verified
//
#include <hip/hip_runtime.h>
#include <hip/hip_bf16.h>
#include <math.h>

typedef __attribute__((ext_vector_type(2))) float v2f;
typedef __attribute__((ext_vector_type(8))) float v8f;

static constexpr int kH = 240;
static constexpr int kW = 320;
static constexpr int kNPix = kH * kW;          // 76800
static constexpr int kNCls = 10;
static constexpr int kThreads = 1024;          // 32 waves of 32 (wave32)
static constexpr int kWaves = kThreads / 32;

// Exact full-wave (32-lane) f32 sum using V_WMMA_F32_16X16X4_F32.
// A(16x4): lane m (<16) supplies (M=m,K=0..1)={v,0}; lane m+16 supplies (M=m,K=2..3)={v,0}.
// B = ones => D[m][*] = rowsum(m) = v[m] + v[m+16].
// D vgpr r in lane L holds rowsum(r + 8*(L>=16)); summing the 8 vgprs gives half-sums,
// one shfl_xor(16) completes the total in every lane. Requires EXEC all ones (uniform call site).
__device__ __forceinline__ float wave_sum_wmma(float v) {
    v2f a; a.x = v;    a.y = 0.0f;
    v2f b; b.x = 1.0f; b.y = 1.0f;
    v8f c = {};
    v8f d = __builtin_amdgcn_wmma_f32_16x16x4_f32(
        /*neg_a=*/false, a, /*neg_b=*/false, b,
        /*c_mod=*/(short)0, c, /*reuse_a=*/false, /*reuse_b=*/false);
    float s = d[0] + d[1] + d[2] + d[3] + d[4] + d[5] + d[6] + d[7];
    s += __shfl_xor(s, 16, 32);
    return s;
}

extern "C" __global__ __launch_bounds__(kThreads, 1)
void hough_vote_kernel(const int* __restrict__ label,
                       const float* __restrict__ cmap,
                       float* __restrict__ out)
{
    extern __shared__ unsigned char smem[];
    unsigned int* hist = (unsigned int*)smem;                                 // 307200 B
    unsigned long long* keys = (unsigned long long*)(smem + (size_t)kNPix * 4); // 8192 B
    float* wsum = (float*)(smem + (size_t)kNPix * 4 + (size_t)kThreads * 8);  // 128 B
    float* wcnt = wsum + kWaves;                                              // 128 B
    unsigned int* cmaskCnt = (unsigned int*)(wcnt + kWaves);                  // 4 B

    const int cls  = blockIdx.x;          // 0..9 -> label value cls+1
    const int tid  = threadIdx.x;
    const int lane = tid & 31;
    const int wave = tid >> 5;

    const float* dirx = cmap + (size_t)(3 * cls) * kNPix;
    const float* diry = dirx + kNPix;
    const float* dep  = dirx + 2 * kNPix;

    // ---- Phase 0: zero LDS histogram + counters ----
    for (int i = tid; i < kNPix; i += kThreads) hist[i] = 0u;
    if (tid == 0) *cmaskCnt = 0u;
    __syncthreads();

    // ---- Phase 1: wave-cooperative voting into LDS histogram ----
    unsigned int myCnt = 0;
    for (int base = wave * 32; base < kNPix; base += kThreads) {
        const int pix = base + lane;
        const int lab = label[pix];
        const float sl_own = diry[pix] / dirx[pix];   // slope at this lane's pixel
        const bool mine = (lab == cls + 1);
        unsigned long long m = __ballot(mine);
        if (lane == 0) myCnt += (unsigned int)__popcll(m);
        while (m) {
            const int s = __ffsll(m) - 1;
            m &= (m - 1ull);
            const float sl = __shfl(sl_own, s, 32);   // broadcast source pixel's slope
            const int spix = base + s;
            const float x0 = (float)(spix % kW);
            const float y0 = (float)(spix / kW);
            // lane l covers columns l, l+32, ... : 32 distinct LDS banks (320 % 64 == 0)
            #pragma unroll
            for (int j = 0; j < kW / 32; ++j) {
                const int xp = lane + (j << 5);
                const float fy = rintf(y0 + sl * ((float)xp - x0));  // round-half-even
                if (fy >= 0.0f && fy < (float)kH) {                  // -0.0 passes, like ref
                    atomicAdd(&hist[(int)fy * kW + xp], 1u);
                }
            }
        }
    }
    if (lane == 0 && myCnt) atomicAdd(cmaskCnt, myCnt);
    __syncthreads();

    // ---- Phase 2: argmax with first-index (lowest-bin) tie break ----
    unsigned int bc = 0u;
    int bb = 0x7FFFFFFF;
    for (int i = tid; i < kNPix; i += kThreads) {
        const unsigned int c = hist[i];
        if (c > bc || (c == bc && i < bb)) { bc = c; bb = i; }
    }
    keys[tid] = ((unsigned long long)bc << 32) |
                (unsigned long long)(0x7FFFFFFFu - (unsigned int)bb);
    __syncthreads();
    for (int off = kThreads / 2; off > 0; off >>= 1) {
        if (tid < off) {
            const unsigned long long o = keys[tid + off];
            if (o > keys[tid]) keys[tid] = o;
        }
        __syncthreads();
    }
    const unsigned long long bk = keys[0];
    const unsigned int inlierNum = (unsigned int)(bk >> 32);
    const int cbin = (int)(0x7FFFFFFFu - (unsigned int)(bk & 0xFFFFFFFFull));
    const float cx = (float)(cbin % kW);
    const float cy = (float)(cbin / kW);

    // ---- Phase 3: inlier mask + depth mean (fixed-order, deterministic) ----
    float dsum = 0.0f;
    unsigned int icnt = 0u;
    for (int i = tid; i < kNPix; i += kThreads) {   // exactly 75 iters per thread: uniform
        if (label[i] == cls + 1) {
            const float px = (float)(i % kW);
            const float py = (float)(i / kW);
            const float dx = px - cx;
            const float dy = py - cy;
            const float dn = sqrtf(dx * dx + dy * dy);
            // center pixel: 0/0 -> NaN -> comparison false (matches JAX)
            const float dot = fabsf(dx / dn * dirx[i] + dy / dn * diry[i]);
            if (dot >= 0.9f) { icnt += 1u; dsum += dep[i]; }
        }
    }
    const float wS = wave_sum_wmma(dsum);            // exact f32 wave reduction via WMMA
    const float wC = wave_sum_wmma((float)icnt);     // integer counts < 2^24: exact in f32
    if (lane == 0) { wsum[wave] = wS; wcnt[wave] = wC; }
    __syncthreads();

    if (tid == 0) {
        float tS = 0.0f, tC = 0.0f;
        for (int wv = 0; wv < kWaves; ++wv) { tS += wsum[wv]; tC += wcnt[wv]; }
        const bool trig = (*cmaskCnt >= 500u) && (inlierNum > 500u);
        out[2 * cls + 0] = trig ? cx : 0.0f;
        out[2 * cls + 1] = trig ? cy : 0.0f;
        const float dmean = (tC > 0.0f) ? (tS / fmaxf(tC, 1.0f))
                                        : __int_as_float(0x7FC00000); // NaN
        out[2 * kNCls + cls] = trig ? dmean : 0.0f;
    }
}

extern "C" void kernel_launch(void* const* d_in, const int* in_sizes, int n_in,
                              void* d_out, int out_size, void* d_ws, size_t ws_size,
                              hipStream_t stream) {
    (void)in_sizes; (void)n_in; (void)out_size; (void)d_ws; (void)ws_size;
    const int*   label = (const int*)d_in[0];    // (1,240,320) int32
    const float* cmap  = (const float*)d_in[1];  // (1,30,240,320) float32
    float* out = (float*)d_out;                  // 20 center floats + 10 depth floats

    const size_t shmem = (size_t)kNPix * 4      // histogram: 307200 B (needs CDNA5 320KB LDS)
                       + (size_t)kThreads * 8   // argmax keys: 8192 B
                       + (size_t)kWaves * 4 * 2 // wave partial sums/counts
                       + 64;                    // counter + pad
    hipLaunchKernelGGL(hough_vote_kernel, dim3(kNCls), dim3(kThreads), shmem, stream,
                       label, cmap, out);
}